// TwoLayerGCN_11897059410286
// MI455X (gfx1250) — compile-verified
//
#include <hip/hip_runtime.h>

typedef float v2f __attribute__((ext_vector_type(2)));
typedef float v8f __attribute__((ext_vector_type(8)));

#define N_NODES  50000
#define N_EDGES  800000
#define F_IN     128
#define HIDDEN   256
#define N_GRAPHS 128

#define KC         64            // K chunk staged in LDS per iteration
#define BN         64            // N columns per block (4 WMMA tiles / wave)
#define LDS_STRIDE (KC + 4)      // pad 4 floats: half-waves hit disjoint bank sets

// ---------------------------------------------------------------------------
// C = A @ B ; A[M,K], B[K,N], C[M,N] row-major; M%16==0, K%KC==0, N%BN==0.
// 256 threads = 8 waves. Each wave: one 16-row M tile x 64 cols (4 f32-WMMA
// accumulators). B is staged in LDS transposed ([n][k], one ds_load_b64 per
// operand); A chunk is preloaded to registers before the staging barrier so
// 16 global_load_b64 are in flight while B staging runs.
// ---------------------------------------------------------------------------
__global__ __launch_bounds__(256) void gemm_f32_wmma(
    const float* __restrict__ A, const float* __restrict__ B,
    float* __restrict__ C, int M, int K, int N)
{
    __shared__ float Bs[BN * LDS_STRIDE];   // 17,408 bytes

    const int tid  = threadIdx.x;
    const int lane = tid & 31;
    const int wave = tid >> 5;
    const int half = lane >> 4;   // 0: lanes 0-15, 1: lanes 16-31
    const int l    = lane & 15;

    const int  mTile  = blockIdx.x * 8 + wave;
    const bool active = (mTile * 16 < M);      // wave-uniform predicate
    const int  nBase  = blockIdx.y * BN;
    const int  m      = active ? (mTile * 16 + l) : 0;
    const float* Arow = A + (size_t)m * K + 2 * half;

    // staging coords: thread covers column sn, k rows sk + 4*i
    const int sn = tid & 63;
    const int sk = tid >> 6;
    const float* Bg = B + (size_t)nBase + sn;

    v8f acc0 = {}, acc1 = {}, acc2 = {}, acc3 = {};

    for (int kbase = 0; kbase < K; kbase += KC) {
        // 1) per-wave A chunk -> registers (loads overlap the B staging below)
        v2f areg[KC / 4];
        if (active) {
            #pragma unroll
            for (int i = 0; i < KC / 4; ++i)
                areg[i] = *(const v2f*)(Arow + kbase + 4 * i);
        }

        // 2) cooperative stage: Bs[n][k] = B[kbase+k][nBase+n] (coalesced reads)
        #pragma unroll
        for (int i = 0; i < KC / 4; ++i) {
            const int k = sk + 4 * i;
            Bs[sn * LDS_STRIDE + k] = Bg[(size_t)(kbase + k) * N];
        }
        __syncthreads();

        // 3) 16 k-steps x 4 N tiles of f32 WMMA, B from LDS
        if (active) {
            #pragma unroll
            for (int i = 0; i < KC / 4; ++i) {
                const int kl = 4 * i + 2 * half;
                v2f b;
                b = *(const v2f*)&Bs[(l +  0) * LDS_STRIDE + kl];
                acc0 = __builtin_amdgcn_wmma_f32_16x16x4_f32(false, areg[i], false, b, (short)0, acc0, false, false);
                b = *(const v2f*)&Bs[(l + 16) * LDS_STRIDE + kl];
                acc1 = __builtin_amdgcn_wmma_f32_16x16x4_f32(false, areg[i], false, b, (short)0, acc1, false, false);
                b = *(const v2f*)&Bs[(l + 32) * LDS_STRIDE + kl];
                acc2 = __builtin_amdgcn_wmma_f32_16x16x4_f32(false, areg[i], false, b, (short)0, acc2, false, false);
                b = *(const v2f*)&Bs[(l + 48) * LDS_STRIDE + kl];
                acc3 = __builtin_amdgcn_wmma_f32_16x16x4_f32(false, areg[i], false, b, (short)0, acc3, false, false);
            }
        }
        __syncthreads();
    }

    // D layout: VGPR r -> tile row (r + 8*half), col l (+16 per N sub-tile)
    if (active) {
        #pragma unroll
        for (int r = 0; r < 8; ++r) {
            float* Crow = C + (size_t)(mTile * 16 + r + 8 * half) * N + nBase + l;
            Crow[0]  = acc0[r];
            Crow[16] = acc1[r];
            Crow[32] = acc2[r];
            Crow[48] = acc3[r];
        }
    }
}

// ---------------------------------------------------------------------------
// SpMM: Y[dst] += w * X[src] over the edge list. 64 threads per edge,
// float4 gathers (X is 51 MB < 192 MB L2), f32 atomic scatter.
// ---------------------------------------------------------------------------
__global__ __launch_bounds__(256) void spmm_edges(
    const float* __restrict__ X, const int* __restrict__ src,
    const int* __restrict__ dst, const float* __restrict__ w,
    float* __restrict__ Y, int nEdges)
{
    const int t = blockIdx.x * 256 + threadIdx.x;
    const int e = t >> 6;
    if (e >= nEdges) return;
    const int f = (t & 63) << 2;

    const int   s  = src[e];
    const int   d  = dst[e];
    const float ww = w[e];

    const float4 v = *(const float4*)(X + (size_t)s * HIDDEN + f);
    float* yp = Y + (size_t)d * HIDDEN + f;
    atomicAdd(yp + 0, v.x * ww);
    atomicAdd(yp + 1, v.y * ww);
    atomicAdd(yp + 2, v.z * ww);
    atomicAdd(yp + 3, v.w * ww);
}

// ---------------------------------------------------------------------------
// Y = relu(Y + bias[col]) elementwise over [rows, HIDDEN]
// ---------------------------------------------------------------------------
__global__ __launch_bounds__(256) void bias_relu(
    float* __restrict__ Y, const float* __restrict__ b, int rows)
{
    const int i = blockIdx.x * 256 + threadIdx.x;
    if (i >= rows * HIDDEN) return;
    const float v = Y[i] + b[i & (HIDDEN - 1)];
    Y[i] = v > 0.f ? v : 0.f;
}

// ---------------------------------------------------------------------------
// Segment pool: G[seg[n]] += H[n]. 64 threads per node row, atomic scatter
// into a 128 KB accumulator (L2-resident).
// ---------------------------------------------------------------------------
__global__ __launch_bounds__(256) void seg_pool(
    const float* __restrict__ H, const int* __restrict__ seg,
    float* __restrict__ G, int nNodes)
{
    const int t = blockIdx.x * 256 + threadIdx.x;
    const int n = t >> 6;
    if (n >= nNodes) return;
    const int f = (t & 63) << 2;

    const int g = seg[n];
    const float4 v = *(const float4*)(H + (size_t)n * HIDDEN + f);
    float* gp = G + (size_t)g * HIDDEN + f;
    atomicAdd(gp + 0, v.x);
    atomicAdd(gp + 1, v.y);
    atomicAdd(gp + 2, v.z);
    atomicAdd(gp + 3, v.w);
}

// ---------------------------------------------------------------------------
// Dense head: out[row] = relu(G[row] @ Wd + bd) @ Wo + bo.
// One block per graph row; LDS tree reduction for the final dot.
// ---------------------------------------------------------------------------
__global__ __launch_bounds__(256) void head_kernel(
    const float* __restrict__ G,  const float* __restrict__ Wd,
    const float* __restrict__ bd, const float* __restrict__ Wo,
    const float* __restrict__ bo, float* __restrict__ out)
{
    __shared__ float grow[HIDDEN];
    __shared__ float red[HIDDEN];

    const int row = blockIdx.x;
    const int j   = threadIdx.x;

    grow[j] = G[(size_t)row * HIDDEN + j];
    __syncthreads();

    float acc = bd[j];
    for (int k = 0; k < HIDDEN; ++k)
        acc = fmaf(grow[k], Wd[(size_t)k * HIDDEN + j], acc);
    acc = acc > 0.f ? acc : 0.f;

    red[j] = acc * Wo[j];
    __syncthreads();
    for (int s = HIDDEN / 2; s > 0; s >>= 1) {
        if (j < s) red[j] += red[j + s];
        __syncthreads();
    }
    if (j == 0) out[row] = red[0] + bo[0];
}

// ---------------------------------------------------------------------------
extern "C" void kernel_launch(void* const* d_in, const int* in_sizes, int n_in,
                              void* d_out, int out_size, void* d_ws, size_t ws_size,
                              hipStream_t stream)
{
    const float* x    = (const float*)d_in[0];
    const int*   esrc = (const int*)  d_in[1];
    const int*   edst = (const int*)  d_in[2];
    const float* ew   = (const float*)d_in[3];
    const int*   seg  = (const int*)  d_in[4];
    const float* W1   = (const float*)d_in[5];
    const float* b1   = (const float*)d_in[6];
    const float* W2   = (const float*)d_in[7];
    const float* b2   = (const float*)d_in[8];
    const float* Wd   = (const float*)d_in[9];
    const float* bd   = (const float*)d_in[10];
    const float* Wo   = (const float*)d_in[11];
    const float* bo   = (const float*)d_in[12];
    float* out = (float*)d_out;

    const size_t NM = (size_t)N_NODES * HIDDEN;     // 12.8M floats
    float* bufA = (float*)d_ws;                     // GEMM outputs (t1, t2)
    float* bufB = bufA + NM;                        // SpMM outputs (s1/h1, s2/h2)
    float* gbuf = bufB + NM;                        // [N_GRAPHS, HIDDEN] pool

    const dim3 blk(256);
    const int  mTiles     = N_NODES / 16;                 // 3125 (exact)
    const dim3 gemmGrid((mTiles + 7) / 8, HIDDEN / BN);   // (391, 4)
    const int  spmmBlocks = (N_EDGES * 64) / 256;         // 200000
    const int  ewBlocks   = (N_NODES * HIDDEN + 255) / 256;
    const int  poolBlocks = (N_NODES * 64) / 256;         // 12500

    // Layer 1: t1 = x @ W1
    gemm_f32_wmma<<<gemmGrid, blk, 0, stream>>>(x, W1, bufA, N_NODES, F_IN, HIDDEN);
    // s1 = A_w @ t1 ; h1 = relu(s1 + b1)
    hipMemsetAsync(bufB, 0, NM * sizeof(float), stream);
    spmm_edges<<<spmmBlocks, blk, 0, stream>>>(bufA, esrc, edst, ew, bufB, N_EDGES);
    bias_relu<<<ewBlocks, blk, 0, stream>>>(bufB, b1, N_NODES);

    // Layer 2: t2 = h1 @ W2 (bufB -> bufA)
    gemm_f32_wmma<<<gemmGrid, blk, 0, stream>>>(bufB, W2, bufA, N_NODES, HIDDEN, HIDDEN);
    // s2 = A_w @ t2 ; h2 = relu(s2 + b2)  (h1 dead -> reuse bufB)
    hipMemsetAsync(bufB, 0, NM * sizeof(float), stream);
    spmm_edges<<<spmmBlocks, blk, 0, stream>>>(bufA, esrc, edst, ew, bufB, N_EDGES);
    bias_relu<<<ewBlocks, blk, 0, stream>>>(bufB, b2, N_NODES);

    // Global sum pool by graph id
    hipMemsetAsync(gbuf, 0, (size_t)N_GRAPHS * HIDDEN * sizeof(float), stream);
    seg_pool<<<poolBlocks, blk, 0, stream>>>(bufB, seg, gbuf, N_NODES);

    // Dense head
    head_kernel<<<N_GRAPHS, blk, 0, stream>>>(gbuf, Wd, bd, Wo, bo, out);
}